// SparseSwiGLU_62380105007473
// MI455X (gfx1250) — compile-verified
//
#include <hip/hip_runtime.h>
#include <math.h>

#define DIM     1024
#define HIDDEN  4096
#define TOKENS  512
#define COO_NNZ 262144

#define KC      128        // K-chunk staged in LDS
#define MT      4          // M-tiles (of 16) per wave -> 4x B reuse
#define WPB     8          // waves per 256-thread block (wave32)
#define M_ROWS  (MT * 16)  // 64 rows of A staged per block

typedef float v2f __attribute__((ext_vector_type(2)));
typedef float v8f __attribute__((ext_vector_type(8)));
typedef __attribute__((address_space(3))) float lds_float;

// A tile in LDS: MT sub-tiles of 16 rows, each with the pair-interleaved layout
//   idx(mt, ml, kk) = mt*16*KC + (kk>>1)*32 + ml*2 + (kk&1)
// so the WMMA A-fragment (lane l: ml = l&15, K = k0 + 2*(l>>4) + {0,1}) is one
// conflict-free b64 load (lanes 0-15 -> banks 0..31, lanes 16-31 -> 32..63).
__device__ __forceinline__ int a_idx(int mt, int ml, int kk) {
  return mt * 16 * KC + ((kk >> 1) << 5) + (ml << 1) + (kk & 1);
}

// CDNA5 async copy global->LDS, tracked on ASYNCcnt (no VGPR round-trip).
// The LDS destination is passed as a real addrspace(3) pointer so (a) the
// shared array escapes (keeps the ds_load reads alive) and (b) the VGPR holds
// the 32-bit LDS byte offset that VDST expects.
__device__ __forceinline__ void stage_async_b64(const float* gsrc, lds_float* ldst) {
  asm volatile("global_load_async_to_lds_b64 %0, %1, off"
               :: "v"(ldst), "v"(gsrc)
               : "memory");
}

__device__ __forceinline__ void wait_async() {
  asm volatile("s_wait_asynccnt 0x0" ::: "memory");
}

__global__ void coo_scatter(const int* __restrict__ row, const int* __restrict__ col,
                            const float* __restrict__ val, float* __restrict__ W,
                            int nnz, int ld) {
  int i = blockIdx.x * blockDim.x + threadIdx.x;
  if (i < nnz) {
    atomicAdd(&W[(size_t)row[i] * (size_t)ld + (size_t)col[i]], val[i]);
  }
}

// hidden = silu(x*Wup^T + bup) * (x*Wgate^T + bgate)
// M=TOKENS, N=HIDDEN, K=DIM. Wave: 1 N-tile x MT M-tiles, 2 accumulator sets.
__global__ __launch_bounds__(256) void upgate_gemm(
    const float* __restrict__ X,      // [TOKENS, DIM]
    const float* __restrict__ Wup,    // [HIDDEN, DIM] dense
    const float* __restrict__ Wgt,    // [HIDDEN, DIM] dense
    const float* __restrict__ bup,
    const float* __restrict__ bgt,
    float* __restrict__ H)            // [TOKENS, HIDDEN]
{
  __shared__ float lds[M_ROWS * KC];  // 32 KB
  const int tid    = threadIdx.x;
  const int lane   = tid & 31;
  const int wave   = tid >> 5;
  const int h      = lane >> 4;       // half-wave: K pair select within K4 step
  const int ml     = lane & 15;       // m within sub-tile / n for B and C/D
  const int m_base = blockIdx.y * M_ROWS;
  const int n_base = (blockIdx.x * WPB + wave) * 16;

  v8f acc_u[MT], acc_g[MT];
  #pragma unroll
  for (int mt = 0; mt < MT; ++mt) {
    acc_u[mt] = (v8f){0.f,0.f,0.f,0.f,0.f,0.f,0.f,0.f};
    acc_g[mt] = (v8f){0.f,0.f,0.f,0.f,0.f,0.f,0.f,0.f};
  }

  const float* wu = Wup + (size_t)(n_base + ml) * DIM;
  const float* wg = Wgt + (size_t)(n_base + ml) * DIM;

  for (int kc = 0; kc < DIM; kc += KC) {
    __syncthreads();
    // Stage A tile: 64 rows x KC floats as (k,k+1) pairs via async b64.
    for (int i = tid; i < M_ROWS * KC / 2; i += 256) {
      int m = i >> 6;                 // / (KC/2)
      int p = i & (KC / 2 - 1);       // pair index within row
      stage_async_b64(X + (size_t)(m_base + m) * DIM + kc + 2 * p,
                      (lds_float*)&lds[a_idx(m >> 4, m & 15, 2 * p)]);
    }
    wait_async();
    __syncthreads();
    if (kc + KC < DIM) {
      __builtin_prefetch(wu + kc + KC, 0, 1);
      __builtin_prefetch(wg + kc + KC, 0, 1);
    }
    #pragma unroll 4
    for (int k0 = 0; k0 < KC; k0 += 4) {
      v2f bu = *(const v2f*)(wu + kc + k0 + 2 * h);
      v2f bg = *(const v2f*)(wg + kc + k0 + 2 * h);
      #pragma unroll
      for (int mt = 0; mt < MT; ++mt) {
        v2f a = *(const v2f*)&lds[a_idx(mt, ml, k0 + 2 * h)];
        acc_u[mt] = __builtin_amdgcn_wmma_f32_16x16x4_f32(false, a, false, bu,
                                                          (short)0, acc_u[mt], false, false);
        acc_g[mt] = __builtin_amdgcn_wmma_f32_16x16x4_f32(false, a, false, bg,
                                                          (short)0, acc_g[mt], false, false);
      }
    }
  }

  const int   feat = n_base + ml;
  const float biu  = bup[feat];
  const float big  = bgt[feat];
  #pragma unroll
  for (int mt = 0; mt < MT; ++mt) {
    #pragma unroll
    for (int v = 0; v < 8; ++v) {
      int   tok = m_base + mt * 16 + v + 8 * h;   // C/D: M = v + 8*(lane>=16)
      float u   = acc_u[mt][v] + biu;
      float g   = acc_g[mt][v] + big;
      float s   = u / (1.0f + __expf(-u));        // silu
      H[(size_t)tok * HIDDEN + feat] = s * g;
    }
  }
}

// out = x + hidden*Wdown^T + bdown.  M=TOKENS, N=DIM, K=HIDDEN.
__global__ __launch_bounds__(256) void down_gemm(
    const float* __restrict__ H,      // [TOKENS, HIDDEN]
    const float* __restrict__ Wd,     // [DIM, HIDDEN] dense
    const float* __restrict__ bd,
    const float* __restrict__ X,      // [TOKENS, DIM] residual
    float* __restrict__ Out)          // [TOKENS, DIM]
{
  __shared__ float lds[M_ROWS * KC];
  const int tid    = threadIdx.x;
  const int lane   = tid & 31;
  const int wave   = tid >> 5;
  const int h      = lane >> 4;
  const int ml     = lane & 15;
  const int m_base = blockIdx.y * M_ROWS;
  const int n_base = (blockIdx.x * WPB + wave) * 16;

  v8f acc[MT];
  #pragma unroll
  for (int mt = 0; mt < MT; ++mt) {
    acc[mt] = (v8f){0.f,0.f,0.f,0.f,0.f,0.f,0.f,0.f};
  }

  const float* wd = Wd + (size_t)(n_base + ml) * HIDDEN;

  for (int kc = 0; kc < HIDDEN; kc += KC) {
    __syncthreads();
    for (int i = tid; i < M_ROWS * KC / 2; i += 256) {
      int m = i >> 6;
      int p = i & (KC / 2 - 1);
      stage_async_b64(H + (size_t)(m_base + m) * HIDDEN + kc + 2 * p,
                      (lds_float*)&lds[a_idx(m >> 4, m & 15, 2 * p)]);
    }
    wait_async();
    __syncthreads();
    if (kc + KC < HIDDEN) {
      __builtin_prefetch(wd + kc + KC, 0, 1);
    }
    #pragma unroll 4
    for (int k0 = 0; k0 < KC; k0 += 4) {
      v2f b = *(const v2f*)(wd + kc + k0 + 2 * h);
      #pragma unroll
      for (int mt = 0; mt < MT; ++mt) {
        v2f a = *(const v2f*)&lds[a_idx(mt, ml, k0 + 2 * h)];
        acc[mt] = __builtin_amdgcn_wmma_f32_16x16x4_f32(false, a, false, b,
                                                        (short)0, acc[mt], false, false);
      }
    }
  }

  const int   feat = n_base + ml;
  const float bias = bd[feat];
  #pragma unroll
  for (int mt = 0; mt < MT; ++mt) {
    #pragma unroll
    for (int v = 0; v < 8; ++v) {
      int tok = m_base + mt * 16 + v + 8 * h;
      size_t o = (size_t)tok * DIM + feat;
      Out[o] = X[o] + acc[mt][v] + bias;
    }
  }
}

extern "C" void kernel_launch(void* const* d_in, const int* in_sizes, int n_in,
                              void* d_out, int out_size, void* d_ws, size_t ws_size,
                              hipStream_t stream) {
  const float* x         = (const float*)d_in[0];
  const int*   up_row    = (const int*)d_in[1];
  const int*   up_col    = (const int*)d_in[2];
  const float* up_val    = (const float*)d_in[3];
  const float* up_bias   = (const float*)d_in[4];
  const int*   gate_row  = (const int*)d_in[5];
  const int*   gate_col  = (const int*)d_in[6];
  const float* gate_val  = (const float*)d_in[7];
  const float* gate_bias = (const float*)d_in[8];
  const int*   down_row  = (const int*)d_in[9];
  const int*   down_col  = (const int*)d_in[10];
  const float* down_val  = (const float*)d_in[11];
  const float* down_bias = (const float*)d_in[12];

  // Workspace layout (f32): Wup 16MB | Wgate 16MB | Wdown 16MB | hidden 8MB = 56MB
  float* Wup  = (float*)d_ws;
  float* Wgt  = Wup + (size_t)HIDDEN * DIM;
  float* Wdn  = Wgt + (size_t)HIDDEN * DIM;
  float* Hbuf = Wdn + (size_t)DIM * HIDDEN;

  hipMemsetAsync(d_ws, 0, 3u * sizeof(float) * (size_t)HIDDEN * (size_t)DIM, stream);

  const int st = 256;
  const int sg = (COO_NNZ + st - 1) / st;
  coo_scatter<<<sg, st, 0, stream>>>(up_row,   up_col,   up_val,   Wup, COO_NNZ, DIM);
  coo_scatter<<<sg, st, 0, stream>>>(gate_row, gate_col, gate_val, Wgt, COO_NNZ, DIM);
  coo_scatter<<<sg, st, 0, stream>>>(down_row, down_col, down_val, Wdn, COO_NNZ, HIDDEN);

  dim3 g1(HIDDEN / (16 * WPB), TOKENS / M_ROWS);   // 32 x 8 blocks
  upgate_gemm<<<g1, 256, 0, stream>>>(x, Wup, Wgt, up_bias, gate_bias, Hbuf);

  dim3 g2(DIM / (16 * WPB), TOKENS / M_ROWS);      // 8 x 8 blocks
  down_gemm<<<g2, 256, 0, stream>>>(Hbuf, Wdn, down_bias, x, (float*)d_out);
}